// MaskedReconstructionLoss_18064632447412
// MI455X (gfx1250) — compile-verified
//
#include <hip/hip_runtime.h>
#include <hip/hip_bf16.h>
#include <math.h>

// Problem constants (from reference)
#define BB 32
#define TT 512
#define DD 512
#define MM 4096
#define NNEG 64
#define RBLK 512              // row block for sim panel (8 blocks total)
#define INV_TEMP 10.0f        // 1 / 0.1

// GEMM tiling
#define KC   32               // K panel per LDS stage
#define LDA  36               // padded LDS row stride (floats): 16B-aligned float4 stores,
                              // 36r mod 64 hits 16 distinct banks for the 16-lane column reads

typedef __attribute__((ext_vector_type(2))) float v2f;
typedef __attribute__((ext_vector_type(8))) float v8f;

// ---------------- helpers ----------------
__device__ __forceinline__ unsigned pcg_hash(unsigned x) {
    x = x * 747796405u + 2891336453u;
    unsigned w = ((x >> ((x >> 28u) + 4u)) ^ x) * 277803737u;
    return (w >> 22u) ^ w;
}

__device__ __forceinline__ float wred_max(float v) {
    #pragma unroll
    for (int o = 16; o > 0; o >>= 1) v = fmaxf(v, __shfl_xor(v, o, 32));
    return v;
}
__device__ __forceinline__ float wred_sum(float v) {
    #pragma unroll
    for (int o = 16; o > 0; o >>= 1) v += __shfl_xor(v, o, 32);
    return v;
}

// ---------------- kernel 1: gather + L2 normalize (one wave per row m) ----------------
__global__ void gather_normalize_kernel(const float* __restrict__ pred,
                                        const float* __restrict__ enc,
                                        const long long* __restrict__ ids,
                                        float* __restrict__ pred_n,
                                        float* __restrict__ enc_n) {
    const int wave = (blockIdx.x * blockDim.x + threadIdx.x) >> 5;   // 0..MM-1
    const int lane = threadIdx.x & 31;
    const int row = (int)ids[2 * wave];
    const int col = (int)ids[2 * wave + 1];
    const size_t src = ((size_t)row * TT + col) * DD;
    const float* ps = pred + src;
    const float* es = enc + src;

    float4 bp[4], be[4];
    float sp = 0.f, se = 0.f;
    #pragma unroll
    for (int i = 0; i < 4; i++) {
        const int j = (i * 32 + lane) * 4;                           // D=512 -> 128 float4
        bp[i] = *(const float4*)(ps + j);
        be[i] = *(const float4*)(es + j);
        sp += bp[i].x*bp[i].x + bp[i].y*bp[i].y + bp[i].z*bp[i].z + bp[i].w*bp[i].w;
        se += be[i].x*be[i].x + be[i].y*be[i].y + be[i].z*be[i].z + be[i].w*be[i].w;
    }
    sp = wred_sum(sp);
    se = wred_sum(se);
    const float scp = 1.0f / fmaxf(sqrtf(sp), 1e-12f);
    const float sce = 1.0f / fmaxf(sqrtf(se), 1e-12f);

    float* pd = pred_n + (size_t)wave * DD;
    float* ed = enc_n  + (size_t)wave * DD;
    #pragma unroll
    for (int i = 0; i < 4; i++) {
        const int j = (i * 32 + lane) * 4;
        float4 o;
        o.x = bp[i].x*scp; o.y = bp[i].y*scp; o.z = bp[i].z*scp; o.w = bp[i].w*scp;
        *(float4*)(pd + j) = o;
        o.x = be[i].x*sce; o.y = be[i].y*sce; o.z = be[i].z*sce; o.w = be[i].w*sce;
        *(float4*)(ed + j) = o;
    }
}

// ---------------- kernel 2: LDS-tiled WMMA fp32 GEMM panel  S[rb] = Pn[rb] * En^T ------
// Workgroup (8 waves) computes a 64x64 block. A/B K-panels staged in LDS with coalesced
// float4 loads; WMMAs fed from LDS. Each wave: one 16-row strip x two 16-col tiles
// (A fragment reused by 2 WMMAs per K step).
// f32 WMMA fragment layout (ISA 7.12.2): lanes 0-15 hold K=0,1; lanes 16-31 hold K=2,3.
__global__ void sim_gemm_kernel(const float* __restrict__ pred_n,
                                const float* __restrict__ enc_n,
                                float* __restrict__ sim,
                                int row0base) {
    __shared__ float As[64 * LDA];
    __shared__ float Bs[64 * LDA];

    const int tid  = threadIdx.x;
    const int wave = tid >> 5;
    const int lane = tid & 31;
    const int lr   = lane & 15;
    const int half = lane >> 4;                   // 0: K0/K1   1: K2/K3

    const int nColBlocks = MM / 64;               // 64
    const int brow = blockIdx.x / nColBlocks;     // 0..RBLK/64-1
    const int bcol = blockIdx.x % nColBlocks;     // 0..63
    const int row0 = row0base + brow * 64;
    const int col0 = bcol * 64;

    // wave -> (tile row strip, pair of tile cols)
    const int tr  = wave >> 1;                    // 0..3
    const int tc0 = (wave & 1) * 2;               // 0 or 2

    // staging map: 4 threads per row, 2 float4 each (32 floats per row)
    const int srow = tid >> 2;                    // 0..63
    const int sseg = tid & 3;                     // 0..3

    v8f acc0 = {0.f,0.f,0.f,0.f,0.f,0.f,0.f,0.f};
    v8f acc1 = {0.f,0.f,0.f,0.f,0.f,0.f,0.f,0.f};

    const float* ga = pred_n + (size_t)(row0 + srow) * DD;
    const float* gb = enc_n  + (size_t)(col0 + srow) * DD;

    for (int k0 = 0; k0 < DD; k0 += KC) {
        // ---- stage K-panel: coalesced global float4 loads ----
        const float4 a0 = *(const float4*)(ga + k0 + sseg * 4);
        const float4 a1 = *(const float4*)(ga + k0 + 16 + sseg * 4);
        const float4 b0 = *(const float4*)(gb + k0 + sseg * 4);
        const float4 b1 = *(const float4*)(gb + k0 + 16 + sseg * 4);
        // prefetch next panel while this stage computes
        if (k0 + KC < DD) {
            __builtin_prefetch(ga + k0 + KC, 0, 0);
            __builtin_prefetch(gb + k0 + KC, 0, 0);
        }
        __syncthreads();                           // previous compute done
        *(float4*)(&As[srow * LDA + sseg * 4])      = a0;
        *(float4*)(&As[srow * LDA + 16 + sseg * 4]) = a1;
        *(float4*)(&Bs[srow * LDA + sseg * 4])      = b0;
        *(float4*)(&Bs[srow * LDA + 16 + sseg * 4]) = b1;
        __syncthreads();                           // panel visible

        // ---- compute: 8 K-steps x 2 WMMA (shared A fragment) ----
        #pragma unroll
        for (int kk = 0; kk < KC; kk += 4) {
            const int kl = kk + 2 * half;
            const float2 av  = *(const float2*)(&As[(tr * 16 + lr) * LDA + kl]);
            const float2 bv0 = *(const float2*)(&Bs[(tc0 * 16 + lr) * LDA + kl]);
            const float2 bv1 = *(const float2*)(&Bs[((tc0 + 1) * 16 + lr) * LDA + kl]);
            v2f a;  a.x  = av.x;  a.y  = av.y;
            v2f bb0; bb0.x = bv0.x; bb0.y = bv0.y;
            v2f bb1; bb1.x = bv1.x; bb1.y = bv1.y;
            acc0 = __builtin_amdgcn_wmma_f32_16x16x4_f32(
                false, a, false, bb0, (short)0, acc0, false, false);
            acc1 = __builtin_amdgcn_wmma_f32_16x16x4_f32(
                false, a, false, bb1, (short)0, acc1, false, false);
        }
    }

    // ---- writeback: C/D layout: VGPR v -> M = v (lanes 0-15) / v+8 (lanes 16-31) ----
    const int n0 = col0 + tc0 * 16 + lr;
    #pragma unroll
    for (int v = 0; v < 8; v++) {
        const int mloc = brow * 64 + tr * 16 + v + half * 8;   // local row in panel
        sim[(size_t)mloc * MM + n0]      = acc0[v] * INV_TEMP;
        sim[(size_t)mloc * MM + n0 + 16] = acc1[v] * INV_TEMP;
    }
}

// ---------------- kernel 3: per-row candidate gather + logsumexp (one wave per row) ----
__global__ void contrast_kernel(const float* __restrict__ sim,
                                int rb0,
                                float* __restrict__ loss_p,
                                float* __restrict__ acc_p) {
    const int wave = blockIdx.x * (blockDim.x >> 5) + (threadIdx.x >> 5);  // 0..RBLK-1
    const int lane = threadIdx.x & 31;
    const int r = rb0 + wave;                       // global row
    const float* srow = sim + (size_t)wave * MM;

    float vals[3];
    int   nv = 0;
    float vmax = -1e30f, maxneg = -1e30f;
    for (int c = lane; c < NNEG + 1; c += 32) {     // 65 candidates
        float v;
        if (c == 0) {
            v = srow[r];                            // positive: diagonal entry
        } else {
            unsigned rr = pcg_hash((unsigned)(r * NNEG + (c - 1))) % (unsigned)(MM - 2);
            int s = (int)rr + ((int)rr >= r ? 1 : 0);
            v = srow[s];
            maxneg = fmaxf(maxneg, v);
        }
        vals[nv++] = v;
        vmax = fmaxf(vmax, v);
    }
    vmax   = wred_max(vmax);
    maxneg = wred_max(maxneg);
    float s = 0.f;
    for (int i = 0; i < nv; i++) s += expf(vals[i] - vmax);
    s = wred_sum(s);
    if (lane == 0) {
        const float sim0 = srow[r];
        const float lse  = vmax + logf(s);
        loss_p[r] = lse - sim0;
        acc_p[r]  = (sim0 >= maxneg) ? 1.0f : 0.0f;
    }
}

// ---------------- kernel 4: final mean reduction -> d_out[0]=loss, d_out[1]=acc --------
__global__ void finalize_kernel(const float* __restrict__ loss_p,
                                const float* __restrict__ acc_p,
                                float* __restrict__ out) {
    __shared__ float sl[256];
    __shared__ float sa[256];
    float l = 0.f, a = 0.f;
    for (int i = threadIdx.x; i < MM; i += 256) { l += loss_p[i]; a += acc_p[i]; }
    sl[threadIdx.x] = l; sa[threadIdx.x] = a;
    __syncthreads();
    for (int st = 128; st > 0; st >>= 1) {
        if (threadIdx.x < st) { sl[threadIdx.x] += sl[threadIdx.x + st];
                                sa[threadIdx.x] += sa[threadIdx.x + st]; }
        __syncthreads();
    }
    if (threadIdx.x == 0) {
        out[0] = sl[0] / (float)MM;
        out[1] = sa[0] / (float)MM;
    }
}

// ---------------- launcher ----------------
extern "C" void kernel_launch(void* const* d_in, const int* in_sizes, int n_in,
                              void* d_out, int out_size, void* d_ws, size_t ws_size,
                              hipStream_t stream) {
    (void)in_sizes; (void)n_in; (void)out_size; (void)ws_size;
    const float*     pred = (const float*)d_in[0];
    const float*     enc  = (const float*)d_in[1];
    const long long* ids  = (const long long*)d_in[2];
    float* out = (float*)d_out;

    // workspace layout
    char* ws = (char*)d_ws;
    float* pred_n = (float*)(ws);                                   // 8 MB
    float* enc_n  = (float*)(ws + (size_t)MM * DD * 4);             // 8 MB
    float* sim    = (float*)(ws + (size_t)2 * MM * DD * 4);         // RBLK*MM*4 = 8 MB
    float* loss_p = (float*)(ws + (size_t)2 * MM * DD * 4 + (size_t)RBLK * MM * 4);
    float* acc_p  = loss_p + MM;

    // 1) gather + normalize: one wave per row, 8 waves/block
    gather_normalize_kernel<<<MM / 8, 256, 0, stream>>>(pred, enc, ids, pred_n, enc_n);

    // 2+3) per row-block: dense LDS-tiled WMMA sim panel, then per-row contrastive reduce
    const int blocksPerPanel = (RBLK / 64) * (MM / 64);             // 8 * 64 = 512
    for (int rb = 0; rb < MM / RBLK; rb++) {
        const int rb0 = rb * RBLK;
        sim_gemm_kernel<<<blocksPerPanel, 256, 0, stream>>>(pred_n, enc_n, sim, rb0);
        contrast_kernel<<<RBLK / 8, 256, 0, stream>>>(sim, rb0, loss_p, acc_p);
    }

    // 4) means
    finalize_kernel<<<1, 256, 0, stream>>>(loss_p, acc_p, out);
}